// AttentionBasedEncoder_63625645523586
// MI455X (gfx1250) — compile-verified
//
#include <hip/hip_runtime.h>

// vertices_to_edges: (B,V,C) f32 + (B,E,2) i32 -> (B,C,E) f32
//   out[b][c][e] = 0.5f * (vt[b][edges[b][e][0]][c] + vt[b][edges[b][e][1]][c])
//
// Memory-bound gather (AI ~0.36 FLOP/B, ~610MB total traffic -> ~27us floor
// at 23.3 TB/s). Strategy:
//  - 2D grid: blockIdx.y = batch (scalar base pointers, no div/mod),
//    blockIdx.x tiles E; lanes consecutive in e -> every per-channel b32
//    store is one full 128B cacheline across the wave (coalesced transpose)
//  - per-lane b64 vector loads of the two 248B vertex rows (8B aligned),
//    default RT temporal hint: 149MB of tables stay resident in 192MB L2
//  - NT store hints: the 446MB write-once output stream bypasses L2 residency
//  - edge records staged via gfx1250 async global->LDS + s_wait_asynccnt

#define B_ 4
#define V_ 150000
#define E_ 450000
#define C_ 62
#define THREADS 256

typedef float v2f __attribute__((ext_vector_type(2)));

__global__ __launch_bounds__(THREADS) void v2e_gather_kernel(
    const float* __restrict__ vt,    // (B, V, C)
    const int* __restrict__ edges,   // (B, E, 2)
    float* __restrict__ out)         // (B, C, E)
{
    const int b = blockIdx.y;                       // uniform -> SGPRs
    int e = blockIdx.x * THREADS + threadIdx.x;
    if (e >= E_) e = E_ - 1;  // tail clamp: duplicate lanes redo identical work

    // Per-batch bases: uniform, computed in scalar regs.
    const float* __restrict__ vt_b    = vt    + (size_t)b * V_ * C_;
    const int*   __restrict__ edges_b = edges + (size_t)b * E_ * 2;
    float*       __restrict__ out_b   = out   + (size_t)b * C_ * E_;

    // ---- Stage this lane's edge record (8B) into LDS via the CDNA5 async
    // ---- data path (ASYNCcnt-tracked), then read it back.
    __shared__ int s_edges[THREADS * 2];
    {
        unsigned loff = (unsigned)(unsigned long long)
            (__attribute__((address_space(3))) int*)&s_edges[threadIdx.x * 2];
        const int* gsrc = edges_b + (size_t)e * 2;
        asm volatile("global_load_async_to_lds_b64 %0, %1, off"
                     :: "v"(loff), "v"(gsrc)
                     : "memory");
        // Each lane consumes only its own LDS slot -> per-wave wait suffices.
        asm volatile("s_wait_asynccnt 0" ::: "memory");
    }
    const int v1 = s_edges[threadIdx.x * 2 + 0];
    const int v2 = s_edges[threadIdx.x * 2 + 1];

    // Vertex rows: 248 bytes each, 8-byte aligned -> b64 loads, L2-resident.
    const float* __restrict__ r1 = vt_b + (size_t)(unsigned)v1 * C_;
    const float* __restrict__ r2 = vt_b + (size_t)(unsigned)v2 * C_;

    // Transposed output base for this edge.
    float* __restrict__ op = out_b + (size_t)e;

#pragma unroll
    for (int c = 0; c < C_; c += 2) {
        v2f a = *(const v2f*)(r1 + c);
        v2f d = *(const v2f*)(r2 + c);
        v2f m = (a + d) * 0.5f;
        // Non-temporal: write-once stream, don't evict the vertex tables.
        __builtin_nontemporal_store(m.x, op + (size_t)(c + 0) * E_);
        __builtin_nontemporal_store(m.y, op + (size_t)(c + 1) * E_);
    }
}

extern "C" void kernel_launch(void* const* d_in, const int* in_sizes, int n_in,
                              void* d_out, int out_size, void* d_ws, size_t ws_size,
                              hipStream_t stream) {
    const float* vt    = (const float*)d_in[0];  // (B,V,C) float32
    const int*   edges = (const int*)d_in[1];    // (B,E,2) int32
    float*       out   = (float*)d_out;          // (B,C,E) float32

    const int blocks_x = (E_ + THREADS - 1) / THREADS;  // 1758
    hipLaunchKernelGGL(v2e_gather_kernel, dim3(blocks_x, B_), dim3(THREADS), 0,
                       stream, vt, edges, out);

    (void)in_sizes; (void)n_in; (void)out_size; (void)d_ws; (void)ws_size;
}